// MultiHeadAttentionRel_24893630447763
// MI455X (gfx1250) — compile-verified
//
#include <hip/hip_runtime.h>
#include <cstdint>
#include <cstddef>

// ---------------------------------------------------------------------------
// MHA with relative position bias on MI455X (gfx1250, wave32).
// All matmuls use V_WMMA_F32_16X16X4_F32 (full fp32 precision; the workload is
// HBM-write bound on the attn/scores outputs, so fp32 matrix ops are free).
// GEMM staging uses gfx1250 async global->LDS copies (ASYNCcnt) when the
// toolchain exposes them.
// ---------------------------------------------------------------------------

typedef float v2f __attribute__((ext_vector_type(2)));
typedef float v8f __attribute__((ext_vector_type(8)));
typedef int   v4i __attribute__((ext_vector_type(4)));

#define WMMA_F32X4(a, b, c) \
  __builtin_amdgcn_wmma_f32_16x16x4_f32(false, (a), false, (b), (short)0, (c), false, false)

#if defined(__has_builtin)
#if __has_builtin(__builtin_amdgcn_global_load_async_to_lds_b128)
#define HAVE_ASYNC_LDS 1
#endif
#endif

// 16-byte global -> LDS copy. Async (ASYNCcnt-tracked) on gfx1250 toolchains
// that expose the builtin; falls back to a register round-trip otherwise.
// Builtin prototype (from hipcc diagnostic): param0 = int4 AS1* (global),
// param1 = int4 AS3* (LDS), then imm offset, imm cpol.
__device__ __forceinline__ void copy16_g2l(const float* g, float* l) {
#ifdef HAVE_ASYNC_LDS
  typedef __attribute__((address_space(1))) v4i* gptr_t;
  typedef __attribute__((address_space(3))) v4i* lptr_t;
  __builtin_amdgcn_global_load_async_to_lds_b128((gptr_t)g, (lptr_t)l, 0, 0);
#else
  *(float4*)l = *(const float4*)g;
#endif
}

__device__ __forceinline__ void wait_async_copies() {
#ifdef HAVE_ASYNC_LDS
#if __has_builtin(__builtin_amdgcn_s_wait_asynccnt)
  __builtin_amdgcn_s_wait_asynccnt(0);
#else
  asm volatile("s_wait_asynccnt 0x0" ::: "memory");
#endif
#endif
}

constexpr int NB  = 4;     // batch
constexpr int S   = 1024;  // sequence
constexpr int E   = 1024;  // embed
constexpr int H   = 16;    // heads
constexpr int Dh  = 64;    // head dim
constexpr int MR  = 8;     // max relative distance
constexpr int MTOK = NB * S;  // 4096 rows for projection GEMMs

// ---------------------------------------------------------------------------
// GEMM: Out = X[M,K] * W[N,K]^T + bias[N]      (M=4096, N=K=1024)
// Block: 256 threads = 8 waves; block tile 128x64; wave tile 16x64.
// LDS-staged K-chunks of 32 with padded stride (36) to avoid bank conflicts.
// split_heads=1 writes Out in [B,H,S,Dh] layout, else row-major [M,N].
// ---------------------------------------------------------------------------
__global__ __launch_bounds__(256)
void gemm_bias_kernel(const float* __restrict__ X, const float* __restrict__ W,
                      const float* __restrict__ bias, float* __restrict__ Out,
                      int split_heads)
{
  __shared__ float Xs[128 * 36];
  __shared__ float Ws[64 * 36];

  const int bn   = blockIdx.x & 15;   // 16 N-tiles of 64
  const int bm   = blockIdx.x >> 4;   // 32 M-tiles of 128
  const int tid  = threadIdx.x;
  const int wid  = tid >> 5;
  const int lane = tid & 31;
  const int ln   = lane & 15;
  const int koff = (lane >> 4) << 1;  // A/B K-pair select per ISA layout

  const v8f vzero = {0.f, 0.f, 0.f, 0.f, 0.f, 0.f, 0.f, 0.f};
  v8f acc[4] = {vzero, vzero, vzero, vzero};

  const int rowX0 = bm * 128;
  const int rowW0 = bn * 64;

  for (int kc = 0; kc < E; kc += 32) {
    // stage X chunk 128x32 and W chunk 64x32 (async global->LDS, b128/lane)
    #pragma unroll
    for (int i = 0; i < 4; ++i) {
      int v = tid + i * 256;                  // 0..1023
      int r = v >> 3, c4 = v & 7;
      copy16_g2l(&X[(size_t)(rowX0 + r) * E + kc + c4 * 4], &Xs[r * 36 + c4 * 4]);
    }
    #pragma unroll
    for (int i = 0; i < 2; ++i) {
      int v = tid + i * 256;                  // 0..511
      int r = v >> 3, c4 = v & 7;
      copy16_g2l(&W[(size_t)(rowW0 + r) * E + kc + c4 * 4], &Ws[r * 36 + c4 * 4]);
    }
    if (kc + 32 < E) {   // global_prefetch_b8 of the next K-chunk
      __builtin_prefetch(&X[(size_t)(rowX0 + (tid >> 1)) * E + kc + 32], 0, 3);
    }
    wait_async_copies();
    __syncthreads();

    const int mrow = wid * 16 + ln;
    #pragma unroll
    for (int kk = 0; kk < 8; ++kk) {
      const int kl = kk * 4 + koff;
      v2f a  = *(const v2f*)&Xs[mrow * 36 + kl];
      v2f b0 = *(const v2f*)&Ws[(0 * 16 + ln) * 36 + kl];
      v2f b1 = *(const v2f*)&Ws[(1 * 16 + ln) * 36 + kl];
      v2f b2 = *(const v2f*)&Ws[(2 * 16 + ln) * 36 + kl];
      v2f b3 = *(const v2f*)&Ws[(3 * 16 + ln) * 36 + kl];
      acc[0] = WMMA_F32X4(a, b0, acc[0]);
      acc[1] = WMMA_F32X4(a, b1, acc[1]);
      acc[2] = WMMA_F32X4(a, b2, acc[2]);
      acc[3] = WMMA_F32X4(a, b3, acc[3]);
    }
    __syncthreads();
  }

  // epilogue: C/D layout -> row = tileM + r + 8*(lane>=16), col = tileN + lane%16
  const int q0 = (lane >> 4) << 3;
  #pragma unroll
  for (int j = 0; j < 4; ++j) {
    const int col = bn * 64 + j * 16 + ln;
    const float bv = bias[col];
    #pragma unroll
    for (int r = 0; r < 8; ++r) {
      const int row = bm * 128 + wid * 16 + q0 + r;
      const float val = acc[j][r] + bv;
      if (split_heads) {
        const int b = row >> 10, s = row & 1023;
        const int h = col >> 6,  d = col & 63;
        Out[((size_t)((b * H + h) * S + s)) * Dh + d] = val;
      } else {
        Out[(size_t)row * E + col] = val;
      }
    }
  }
}

// ---------------------------------------------------------------------------
// scores[b,h,q,k] = (Q . K)/sqrt(Dh) + rel_bias[h, clip(k-q)] ; mask -> -inf
// One wave per 16x16 score tile; 16 WMMA steps over Dh=64.
// ---------------------------------------------------------------------------
__global__ __launch_bounds__(256)
void scores_kernel(const float* __restrict__ Q, const float* __restrict__ K,
                   const unsigned char* __restrict__ mask,
                   const float* __restrict__ rel_bias,
                   float* __restrict__ scores)
{
  const int lane = threadIdx.x & 31;
  const int ln   = lane & 15;
  const int koff = (lane >> 4) << 1;
  const int wg   = blockIdx.x * 8 + (threadIdx.x >> 5);   // 0..262143
  const int kt = wg & 63;
  const int qt = (wg >> 6) & 63;
  const int h  = (wg >> 12) & 15;
  const int b  = wg >> 16;

  const float* Qp = Q + (size_t)((b * H + h) * S) * Dh;
  const float* Kp = K + (size_t)((b * H + h) * S) * Dh;

  v8f acc = {0.f, 0.f, 0.f, 0.f, 0.f, 0.f, 0.f, 0.f};
  const int qm = qt * 16 + ln;   // A row
  const int kn = kt * 16 + ln;   // B column (K row index)
  #pragma unroll
  for (int d4 = 0; d4 < 16; ++d4) {
    const int kd = d4 * 4 + koff;
    v2f a  = *(const v2f*)&Qp[(size_t)qm * Dh + kd];
    v2f bb = *(const v2f*)&Kp[(size_t)kn * Dh + kd];
    acc = WMMA_F32X4(a, bb, acc);
  }

  const int q0  = (lane >> 4) << 3;
  const int col = kt * 16 + ln;
  float* Srow = scores + (size_t)((b * H + h) * S) * S;
  const unsigned char* Mrow = mask + (size_t)b * S * S;
  #pragma unroll
  for (int r = 0; r < 8; ++r) {
    const int q = qt * 16 + q0 + r;
    int rel = col - q;
    rel = rel < -MR ? -MR : (rel > MR ? MR : rel);
    float val = acc[r] * 0.125f + rel_bias[h * (2 * MR + 1) + rel + MR];
    if (Mrow[(size_t)q * S + col]) val = -__builtin_inff();
    Srow[(size_t)q * S + col] = val;
  }
}

// ---------------------------------------------------------------------------
// Row softmax over S=1024: one 256-thread block per row (float4 per thread).
// ---------------------------------------------------------------------------
__global__ __launch_bounds__(256)
void softmax_kernel(const float* __restrict__ scores, float* __restrict__ attn)
{
  __shared__ float red[8];
  __shared__ float bc;
  const int tid = threadIdx.x;
  const int wid = tid >> 5, lane = tid & 31;
  const size_t base = (size_t)blockIdx.x * S;

  float4 x = ((const float4*)(scores + base))[tid];
  float m = fmaxf(fmaxf(x.x, x.y), fmaxf(x.z, x.w));
  #pragma unroll
  for (int off = 16; off > 0; off >>= 1)
    m = fmaxf(m, __shfl_xor(m, off, 32));
  if (lane == 0) red[wid] = m;
  __syncthreads();
  if (tid == 0) {
    float mm = red[0];
    #pragma unroll
    for (int i = 1; i < 8; ++i) mm = fmaxf(mm, red[i]);
    bc = mm;
  }
  __syncthreads();
  m = bc;

  float4 e;
  e.x = __expf(x.x - m); e.y = __expf(x.y - m);
  e.z = __expf(x.z - m); e.w = __expf(x.w - m);
  float s = e.x + e.y + e.z + e.w;
  #pragma unroll
  for (int off = 16; off > 0; off >>= 1)
    s += __shfl_xor(s, off, 32);
  if (lane == 0) red[wid] = s;
  __syncthreads();
  if (tid == 0) {
    float ss = 0.f;
    #pragma unroll
    for (int i = 0; i < 8; ++i) ss += red[i];
    bc = ss;
  }
  __syncthreads();
  const float inv = 1.0f / bc;

  float4 o = {e.x * inv, e.y * inv, e.z * inv, e.w * inv};
  ((float4*)(attn + base))[tid] = o;
}

// ---------------------------------------------------------------------------
// ctx = attn @ V ; written transposed into [B,S,E] so out-proj is plain GEMM.
// One wave per (b,h,qtile) -> 16x64 output, K loop over S=1024.
// ---------------------------------------------------------------------------
__global__ __launch_bounds__(256)
void ctx_kernel(const float* __restrict__ attn, const float* __restrict__ V,
                float* __restrict__ ctx)
{
  const int lane = threadIdx.x & 31;
  const int ln   = lane & 15;
  const int koff = (lane >> 4) << 1;
  const int wg   = blockIdx.x * 8 + (threadIdx.x >> 5);   // 0..4095
  const int qt = wg & 63;
  const int h  = (wg >> 6) & 15;
  const int b  = wg >> 10;

  const float* Ap = attn + (size_t)((b * H + h) * S) * S;
  const float* Vp = V + (size_t)((b * H + h) * S) * Dh;

  const v8f vzero = {0.f, 0.f, 0.f, 0.f, 0.f, 0.f, 0.f, 0.f};
  v8f acc[4] = {vzero, vzero, vzero, vzero};
  const int qm = qt * 16 + ln;

  #pragma unroll 4
  for (int ks = 0; ks < S / 4; ++ks) {
    const int kk = ks * 4 + koff;
    v2f a = *(const v2f*)&Ap[(size_t)qm * S + kk];
    #pragma unroll
    for (int j = 0; j < 4; ++j) {
      const int n = j * 16 + ln;
      v2f bb;
      bb.x = Vp[(size_t)kk * Dh + n];
      bb.y = Vp[(size_t)(kk + 1) * Dh + n];
      acc[j] = WMMA_F32X4(a, bb, acc[j]);
    }
  }

  const int q0 = (lane >> 4) << 3;
  #pragma unroll
  for (int j = 0; j < 4; ++j) {
    const int d = j * 16 + ln;
    #pragma unroll
    for (int r = 0; r < 8; ++r) {
      const int q = qt * 16 + q0 + r;
      ctx[((size_t)(b * S + q)) * E + h * Dh + d] = acc[j][r];
    }
  }
}

// ---------------------------------------------------------------------------
extern "C" void kernel_launch(void* const* d_in, const int* in_sizes, int n_in,
                              void* d_out, int out_size, void* d_ws, size_t ws_size,
                              hipStream_t stream) {
  const float*         query = (const float*)d_in[0];
  const float*         key   = (const float*)d_in[1];
  const float*         value = (const float*)d_in[2];
  const unsigned char* mask  = (const unsigned char*)d_in[3];
  const float* Wq = (const float*)d_in[4];  const float* bq = (const float*)d_in[5];
  const float* Wk = (const float*)d_in[6];  const float* bk = (const float*)d_in[7];
  const float* Wv = (const float*)d_in[8];  const float* bv = (const float*)d_in[9];
  const float* Wo = (const float*)d_in[10]; const float* bo = (const float*)d_in[11];
  const float* rb = (const float*)d_in[12];

  // d_out = [out (B*S*E) | attn (B*H*S*S) | scores (B*H*S*S)]
  float* out  = (float*)d_out;
  float* attn = out + (size_t)NB * S * E;
  float* scr  = attn + (size_t)NB * H * S * S;

  // workspace: Q,K,V in [B,H,S,Dh] + ctx in [B,S,E]  (4 x 16 MB = 64 MB)
  float* ws = (float*)d_ws;
  const size_t QN = (size_t)NB * H * S * Dh;
  float* Qw = ws;
  float* Kw = ws + QN;
  float* Vw = ws + 2 * QN;
  float* Cw = ws + 3 * QN;

  const dim3 blk(256);
  gemm_bias_kernel<<<(MTOK / 128) * (E / 64), blk, 0, stream>>>(query, Wq, bq, Qw, 1);
  gemm_bias_kernel<<<(MTOK / 128) * (E / 64), blk, 0, stream>>>(key,   Wk, bk, Kw, 1);
  gemm_bias_kernel<<<(MTOK / 128) * (E / 64), blk, 0, stream>>>(value, Wv, bv, Vw, 1);
  scores_kernel<<<(NB * H * (S / 16) * (S / 16)) / 8, blk, 0, stream>>>(Qw, Kw, mask, rb, scr);
  softmax_kernel<<<NB * H * S, blk, 0, stream>>>(scr, attn);
  ctx_kernel<<<(NB * H * (S / 16)) / 8, blk, 0, stream>>>(attn, Vw, Cw);
  gemm_bias_kernel<<<(MTOK / 128) * (E / 64), blk, 0, stream>>>(Cw, Wo, bo, out, 0);
}